// MyModel_87522843560869
// MI455X (gfx1250) — compile-verified
//
#include <hip/hip_runtime.h>

typedef __attribute__((ext_vector_type(2))) float v2f;
typedef __attribute__((ext_vector_type(4))) float v4f;
typedef __attribute__((ext_vector_type(8))) float v8f;

#define NB  8
#define H_  256
#define W_  256
#define C_  64
#define HP  64   // pooled H
#define WP  64   // pooled W
#define LDS_PITCH 68   // 64 + 4 pad: upper half-wave lands 32 banks away -> conflict-free

// ---------------------------------------------------------------------------
// Kernel 1: 6x6 / stride-4 SAME average pool.  pooled[b,ho,wo,c]
// window rows = ho*4-1 .. ho*4+4 (SAME padding lo=1), divide by valid count
// (counts tensor == valid_h * valid_w, matching the reference's ones-reduce).
// ---------------------------------------------------------------------------
__global__ void avgpool_k(const float* __restrict__ in, float* __restrict__ pooled) {
  const int tid = threadIdx.x;
  const int c   = (tid & 15) * 4;                 // 16 float4 channel groups
  const int wo  = blockIdx.x * 16 + (tid >> 4);   // 16 wo per block
  const int ho  = blockIdx.y;
  const int b   = blockIdx.z;

  const int h0  = ho * 4 - 1;
  const int w0  = wo * 4 - 1;
  const int hlo = max(h0, 0),  hhi = min(h0 + 6, H_);
  const int wlo = max(w0, 0),  whi = min(w0 + 6, W_);

  v4f acc = {};
  for (int h = hlo; h < hhi; ++h) {
    const float* rowp = in + ((((size_t)b * H_ + h) * W_ + wlo) * C_ + c);
    for (int w = wlo; w < whi; ++w) {
      v4f v = *reinterpret_cast<const v4f*>(rowp);   // global_load_b128
      acc += v;
      rowp += C_;
    }
  }
  const float cnt = (float)((hhi - hlo) * (whi - wlo));
  v4f r = acc / cnt;
  *reinterpret_cast<v4f*>(pooled + ((((size_t)b * HP + ho) * WP + wo) * C_ + c)) = r;
}

// ---------------------------------------------------------------------------
// Upsample source-coordinate mapping (stride=4, dsi=2, max_source=63):
//   d <  5.5   : s = (d - 2)     / 3.5
//   d >  249.5 : s = (d - 249.5) / 3.5 + 62
//   else       : s = (d - 1.5)   / 4
// Taps: (r0, 1-fr), (r0+1, fr); taps outside [0,63] are zero-masked in the
// reference — we never include them in the 8-row K window.
// ---------------------------------------------------------------------------
__device__ __forceinline__ float src_coord(int d) {
  const float df = (float)d;
  const float s1 = (df - 2.0f)   / 3.5f;
  const float s2 = (df - 249.5f) / 3.5f + 62.0f;
  const float s3 = (df - 1.5f)   / 4.0f;
  return (df < 5.5f) ? s1 : ((df > 249.5f) ? s2 : s3);   // v_cndmask, no EXEC change
}

// ---------------------------------------------------------------------------
// Kernel 2: vertical upsample via WMMA.  tmp[b,hd,wo,c] = sum_r Uh[hd,r]*pooled[b,r,wo,c]
// One wave computes a 16(hd) x 16(c) tile: A(16x8)=Uh window, B(8x16)=pooled rows.
// ISA VGPR layouts (f32 16x16x4):
//   A: lane m=lane&15; VGPR0/1 hold K = 2*half {+1}   (half = lane>>4)
//   B: VGPR0/1 hold rows K = 2*half {+1}, col n = lane&15
//   D: VGPR j -> row j + 8*half, col n
// The 4 waves stage the joint 16(hd) x 64(c) tile in LDS, then re-read it so
// every thread emits two aligned global_store_b128 (full 256B lines per row).
// ---------------------------------------------------------------------------
__global__ void upsample_h_wmma(const float* __restrict__ pooled, float* __restrict__ tmp) {
  __shared__ float lds[16 * LDS_PITCH];

  const int lane = threadIdx.x & 31;
  const int wave = threadIdx.x >> 5;      // 4 waves -> 4 channel tiles
  const int n    = lane & 15;
  const int half = lane >> 4;
  const int c0   = wave * 16;

  const int t    = blockIdx.x;            // hd tile (16 dest rows)
  const int wo   = blockIdx.y;
  const int b    = blockIdx.z;
  const int hd0  = t * 16;
  const int rbase = min(max(4 * t - 2, 0), HP - 8);

  // per-lane A-row state (dest row d = hd0 + m, m = n)
  const int   d  = hd0 + n;
  const float s  = src_coord(d);
  const int   r0 = (int)floorf(s);
  const float fr = s - (float)r0;

  const size_t src_base = (size_t)b * HP * WP * C_ + (size_t)wo * C_ + (c0 + n);
  const size_t rstride  = (size_t)WP * C_;

  v8f acc = {};
#pragma unroll
  for (int kk = 0; kk < 2; ++kk) {
    const int k0 = kk * 4 + 2 * half;     // K index of this lane's first element
    const int ra = rbase + k0;
    const int rb = ra + 1;
    v2f a, bb;
    a.x  = (ra == r0) ? (1.0f - fr) : ((ra == r0 + 1) ? fr : 0.0f);
    a.y  = (rb == r0) ? (1.0f - fr) : ((rb == r0 + 1) ? fr : 0.0f);
    bb.x = pooled[src_base + (size_t)ra * rstride];
    bb.y = pooled[src_base + (size_t)rb * rstride];
    acc = __builtin_amdgcn_wmma_f32_16x16x4_f32(false, a, false, bb,
                                                (short)0, acc, false, false);
  }

  // D tile -> LDS (ds_store_b32, conflict-free with padded pitch)
#pragma unroll
  for (int j = 0; j < 8; ++j) {
    lds[(j + 8 * half) * LDS_PITCH + c0 + n] = acc[j];
  }
  __syncthreads();

  // linear re-read: thread -> (hd = tid>>3, c = (tid&7)*8), two b128 stores
  const int tid = threadIdx.x;
  const int hdl = tid >> 3;
  const int cq  = (tid & 7) * 8;
  const float* lp = &lds[hdl * LDS_PITCH + cq];
  v4f v0 = *reinterpret_cast<const v4f*>(lp);
  v4f v1 = *reinterpret_cast<const v4f*>(lp + 4);
  float* op = tmp + (((size_t)b * H_ + hd0 + hdl) * WP + wo) * C_ + cq;
  *reinterpret_cast<v4f*>(op)     = v0;   // tmp is re-read by kernel 3: keep RT (L2)
  *reinterpret_cast<v4f*>(op + 4) = v1;
}

// ---------------------------------------------------------------------------
// Kernel 3: horizontal upsample via WMMA.  out[b,hd,wd,c] = sum_s Uw[wd,s]*tmp[b,hd,s,c]
// Same tile structure; the joint 16(wd) x 64(c) tile is a CONTIGUOUS 4KB block
// of the output -> linear nontemporal b128 streaming stores after LDS staging.
// ---------------------------------------------------------------------------
__global__ void upsample_w_wmma(const float* __restrict__ tmp, float* __restrict__ out) {
  __shared__ float lds[16 * LDS_PITCH];

  const int lane = threadIdx.x & 31;
  const int wave = threadIdx.x >> 5;
  const int n    = lane & 15;
  const int half = lane >> 4;
  const int c0   = wave * 16;

  const int t    = blockIdx.x;            // wd tile
  const int hd   = blockIdx.y;
  const int b    = blockIdx.z;
  const int wd0  = t * 16;
  const int rbase = min(max(4 * t - 2, 0), WP - 8);

  const int   d  = wd0 + n;
  const float s  = src_coord(d);
  const int   r0 = (int)floorf(s);
  const float fr = s - (float)r0;

  const size_t src_base = ((size_t)b * H_ + hd) * WP * C_ + (c0 + n);

  v8f acc = {};
#pragma unroll
  for (int kk = 0; kk < 2; ++kk) {
    const int k0 = kk * 4 + 2 * half;
    const int ra = rbase + k0;
    const int rb = ra + 1;
    v2f a, bb;
    a.x  = (ra == r0) ? (1.0f - fr) : ((ra == r0 + 1) ? fr : 0.0f);
    a.y  = (rb == r0) ? (1.0f - fr) : ((rb == r0 + 1) ? fr : 0.0f);
    bb.x = tmp[src_base + (size_t)ra * C_];
    bb.y = tmp[src_base + (size_t)rb * C_];
    acc = __builtin_amdgcn_wmma_f32_16x16x4_f32(false, a, false, bb,
                                                (short)0, acc, false, false);
  }

  // D tile -> LDS
#pragma unroll
  for (int j = 0; j < 8; ++j) {
    lds[(j + 8 * half) * LDS_PITCH + c0 + n] = acc[j];
  }
  __syncthreads();

  // tile is contiguous in out: element index = wd*64 + c = tid*8
  const int tid = threadIdx.x;
  const int wdl = tid >> 3;
  const int cq  = (tid & 7) * 8;
  const float* lp = &lds[wdl * LDS_PITCH + cq];
  v4f v0 = *reinterpret_cast<const v4f*>(lp);
  v4f v1 = *reinterpret_cast<const v4f*>(lp + 4);
  float* op = out + (((size_t)b * H_ + hd) * W_ + wd0) * C_ + tid * 8;
  __builtin_nontemporal_store(v0, reinterpret_cast<v4f*>(op));      // streaming b128
  __builtin_nontemporal_store(v1, reinterpret_cast<v4f*>(op + 4));
}

// ---------------------------------------------------------------------------
extern "C" void kernel_launch(void* const* d_in, const int* in_sizes, int n_in,
                              void* d_out, int out_size, void* d_ws, size_t ws_size,
                              hipStream_t stream) {
  (void)in_sizes; (void)n_in; (void)out_size; (void)ws_size;
  const float* in  = (const float*)d_in[0];
  float* out       = (float*)d_out;
  float* pooled    = (float*)d_ws;                             // 8*64*64*64   =  8.4 MB
  float* tmp       = pooled + (size_t)NB * HP * WP * C_;       // 8*256*64*64  = 33.6 MB

  // pool: grid (wo_tiles=4, ho=64, b=8), 256 threads = 16 wo x 16 float4-channel groups
  avgpool_k<<<dim3(4, 64, 8), 256, 0, stream>>>(in, pooled);

  // vertical upsample: grid (hd_tiles=16, wo=64, b=8), 4 waves = 4 channel tiles
  upsample_h_wmma<<<dim3(16, 64, 8), 128, 0, stream>>>(pooled, tmp);

  // horizontal upsample: grid (wd_tiles=16, hd=256, b=8)
  upsample_w_wmma<<<dim3(16, 256, 8), 128, 0, stream>>>(tmp, out);
}